// Model_51453708206346
// MI455X (gfx1250) — compile-verified
//
#include <hip/hip_runtime.h>
#include <hip/hip_bf16.h>

#define BLOCK 256
#define WAVE_SZ 32
#define WAVES_PER_BLOCK (BLOCK / WAVE_SZ)   // 8 waves on wave32
#define MAX_BINS 256

typedef float v4f __attribute__((ext_vector_type(4)));  // clang vector: NT-load legal

// ---------------------------------------------------------------------------
// Kernel 1: zero the u32 accumulator workspace (graph-capture safe re-init).
// ---------------------------------------------------------------------------
__global__ void hist_zero_ws(unsigned int* __restrict__ ws, int bins) {
    int i = blockIdx.x * blockDim.x + threadIdx.x;
    if (i < bins) ws[i] = 0u;
}

// ---------------------------------------------------------------------------
// Kernel 2: streaming histogram.
//   - v4f (global_load_b128) non-temporal loads: data is read exactly once
//     and exceeds the 192 MB L2, so NT avoids cache thrash.
//   - per-wave private LDS histograms (8 x 256 u32 = 8 KB of the 320 KB WGP
//     LDS) -> ds_add_u32 with zero inter-wave contention.
//   - one u32 global atomicAdd per (bin, block) at the end.
// ---------------------------------------------------------------------------
__global__ __launch_bounds__(BLOCK)
void hist_main(const float* __restrict__ x, long long n,
               const int* __restrict__ p_min, const int* __restrict__ p_max,
               int bins, unsigned int* __restrict__ ws) {
    __shared__ unsigned int lhist[WAVES_PER_BLOCK * MAX_BINS];

    // Zero LDS sub-histograms.
    for (int i = threadIdx.x; i < WAVES_PER_BLOCK * MAX_BINS; i += BLOCK)
        lhist[i] = 0u;
    __syncthreads();

    const float lo    = (float)(*p_min);
    const float hi    = (float)(*p_max);
    const float width = (hi - lo) / (float)bins;     // matches reference exactly
    const float inv_w = 1.0f / width;                // exact for pow2 width (0.03125)
    const int   last  = bins - 1;

    unsigned int* myh = &lhist[(threadIdx.x / WAVE_SZ) * MAX_BINS];

    const long long n4     = n >> 2;
    const long long tid    = (long long)blockIdx.x * BLOCK + threadIdx.x;
    const long long stride = (long long)gridDim.x * BLOCK;

    const v4f* __restrict__ x4 = (const v4f*)x;
    for (long long i = tid; i < n4; i += stride) {
        v4f v = __builtin_nontemporal_load(&x4[i]);
        float e[4] = {v.x, v.y, v.z, v.w};
        #pragma unroll
        for (int c = 0; c < 4; ++c) {
            float xv = e[c];
            if (xv >= lo && xv <= hi) {              // out-of-range dropped
                int b = (int)floorf((xv - lo) * inv_w);
                b = b < 0 ? 0 : (b > last ? last : b);   // x == hi -> last bin
                atomicAdd(&myh[b], 1u);              // ds_add_u32
            }
        }
    }
    // Scalar tail (n not multiple of 4).
    const long long tail_base = n4 << 2;
    if (tid < (n - tail_base)) {
        float xv = x[tail_base + tid];
        if (xv >= lo && xv <= hi) {
            int b = (int)floorf((xv - lo) * inv_w);
            b = b < 0 ? 0 : (b > last ? last : b);
            atomicAdd(&myh[b], 1u);
        }
    }
    __syncthreads();

    // Block reduction across the 8 wave-private copies, then one global
    // integer atomic per bin per block (deterministic; u32 counts are exact).
    for (int b = threadIdx.x; b < bins; b += BLOCK) {
        unsigned int s = 0;
        #pragma unroll
        for (int w = 0; w < WAVES_PER_BLOCK; ++w) s += lhist[w * MAX_BINS + b];
        if (s) atomicAdd(&ws[b], s);
    }
}

// ---------------------------------------------------------------------------
// Kernel 3: convert u32 counts -> fp32 output (reference output dtype).
// Writes every output element, so poisoned d_out needs no pre-zeroing.
// ---------------------------------------------------------------------------
__global__ void hist_finalize(const unsigned int* __restrict__ ws,
                              float* __restrict__ out, int bins) {
    int i = blockIdx.x * blockDim.x + threadIdx.x;
    if (i < bins) out[i] = (float)ws[i];
}

extern "C" void kernel_launch(void* const* d_in, const int* in_sizes, int n_in,
                              void* d_out, int out_size, void* d_ws, size_t ws_size,
                              hipStream_t stream) {
    const float* x     = (const float*)d_in[0];
    const int*   p_min = (const int*)d_in[2];
    const int*   p_max = (const int*)d_in[3];
    float*       out   = (float*)d_out;
    unsigned int* ws   = (unsigned int*)d_ws;

    const long long n    = (long long)in_sizes[0];
    const int       bins = out_size;                 // 256

    // Enough blocks to saturate all WGPs; each thread streams ~16 float4s.
    long long n4 = n >> 2;
    int blocks = (int)((n4 + BLOCK - 1) / BLOCK);
    if (blocks > 4096) blocks = 4096;
    if (blocks < 1) blocks = 1;

    hist_zero_ws<<<(bins + BLOCK - 1) / BLOCK, BLOCK, 0, stream>>>(ws, bins);
    hist_main<<<blocks, BLOCK, 0, stream>>>(x, n, p_min, p_max, bins, ws);
    hist_finalize<<<(bins + BLOCK - 1) / BLOCK, BLOCK, 0, stream>>>(ws, out, bins);
}